// MPJPE_45397804319490
// MI455X (gfx1250) — compile-verified
//
#include <hip/hip_runtime.h>

#define BB 32
#define JJ 13
#define BJ (BB * JJ)      // 416
#define HH 256
#define WW 256

typedef float v2f __attribute__((ext_vector_type(2)));
typedef float v4f __attribute__((ext_vector_type(4)));
typedef float v8f __attribute__((ext_vector_type(8)));
typedef unsigned int u32x4 __attribute__((ext_vector_type(4)));
typedef int i32x4 __attribute__((ext_vector_type(4)));
typedef int i32x8 __attribute__((ext_vector_type(8)));

// Single-instruction XOR lane exchange: ds_swizzle_b32, group-of-32 mode,
// and_mask=0x1F, or_mask=0, xor_mask=M  ->  src_lane = lane ^ M.
template <int M>
__device__ __forceinline__ float xor_lane(float v) {
  return __int_as_float(
      __builtin_amdgcn_ds_swizzle(__float_as_int(v), 0x1F | (M << 10)));
}

// Wave32 sum via V_WMMA_F32_16X16X4_F32 (documented A & C/D layouts; B=ones is
// layout-proof).  A VGPR0 = x (lane l -> M=l%16, K=0/2), VGPR1 = 0, C = 0.
// D[m][*] = x_m + x_{m+16}; per-lane sum of 8 D regs = half-sum, one
// ds_swizzle(xor 16) completes the 32-lane sum.
__device__ __forceinline__ float wave_sum32(float x) {
  v2f a; a.x = x;    a.y = 0.0f;
  v2f b; b.x = 1.0f; b.y = 1.0f;
  v8f c = {};
  c = __builtin_amdgcn_wmma_f32_16x16x4_f32(false, a, false, b, (short)0, c,
                                            false, false);
  float h = ((c[0] + c[1]) + (c[2] + c[3])) + ((c[4] + c[5]) + (c[6] + c[7]));
  h += xor_lane<16>(h);
  return h;
}

__device__ __forceinline__ float wave_max32(float x) {
  x = fmaxf(x, xor_lane<16>(x));
  x = fmaxf(x, xor_lane<8>(x));
  x = fmaxf(x, xor_lane<4>(x));
  x = fmaxf(x, xor_lane<2>(x));
  x = fmaxf(x, xor_lane<1>(x));
  return x;
}

// TDM: 1-D tile (nelem f32, contiguous) global -> LDS.  D# per ISA §8:
// group0 = {count=1 | lds_addr | global_addr[56:0] | type=2},
// group1 = {data_size=4B, tensor_dim0=nelem, tensor_dim1=1, tile_dim0=nelem,
//           tile_dim1=1, stride=nelem}.  Groups 2/3 zero (<=2D tensor).
// This toolchain exposes the 6-arg builtin: (g0, g1, g2:i32x4, g3:i32x4,
// extra:i32x8, cpol).
__device__ __forceinline__ void tdm_load_1d(unsigned int lds_off,
                                            unsigned long long gaddr,
                                            unsigned int nelem) {
  u32x4 g0;
  g0[0] = 1u;                                               // count=1 (user D#)
  g0[1] = lds_off;                                          // lds_addr (bytes)
  g0[2] = (unsigned int)(gaddr & 0xFFFFFFFFull);            // global_addr lo
  g0[3] = (unsigned int)((gaddr >> 32) & 0x01FFFFFFull)     // global_addr hi
          | (2u << 30);                                     // type=2 ("image")
  i32x8 g1;
  g1[0] = (int)(2u << 16);                    // data_size=2 (4 bytes)
  g1[1] = (int)(nelem << 16);                 // tensor_dim0[15:0]
  g1[2] = (int)((nelem >> 16) | (1u << 16));  // tensor_dim0[31:16] | tdim1 lo=1
  g1[3] = (int)(nelem << 16);                 // tdim1 hi=0 | tile_dim0=nelem
  g1[4] = 1;                                  // tile_dim1=1, tile_dim2=0
  g1[5] = (int)nelem;                         // tensor_dim0_stride lo
  g1[6] = 0;
  g1[7] = 0;
  i32x4 z4 = {0, 0, 0, 0};
  i32x8 z8 = {0, 0, 0, 0, 0, 0, 0, 0};
  __builtin_amdgcn_tensor_load_to_lds(g0, g1, z4, z4, z8, 0);
}

// -------- Kernel 1: one block per (b,j); per-row stats in a single HBM pass.
// 8 waves x 32 rows, coalesced b128 loads, software-pipelined one row-group
// ahead.  y_pr is loaded non-temporally so y_gt stays L2-resident (192MB L2,
// y_gt = 109MB) for the TDM mask pass that follows.
__global__ void __launch_bounds__(256) k_rowstats(const float* __restrict__ ypr,
                                                  const float* __restrict__ ygt,
                                                  v4f* __restrict__ rowstats) {
  const int bj   = blockIdx.x;
  const int lane = threadIdx.x & 31;
  const int wv   = threadIdx.x >> 5;
  const size_t slab = (size_t)bj * (HH * WW);
  const v4f* pr4 = (const v4f*)(ypr + slab);
  const v4f* gt4 = (const v4f*)(ygt + slab);

  int ro = wv * (WW / 4);
  v4f p0 = __builtin_nontemporal_load(pr4 + ro + lane);
  v4f p1 = __builtin_nontemporal_load(pr4 + ro + 32 + lane);
  v4f g0 = gt4[ro + lane];
  v4f g1 = gt4[ro + 32 + lane];

  for (int it = 0; it < 32; ++it) {
    const int h = wv + (it << 3);
    // pipeline next row-group (wrap keeps the loop branch-free; the final
    // wrapped loads are redundant but harmless)
    const int nro = ((h + 8) & 255) * (WW / 4);
    v4f np0 = __builtin_nontemporal_load(pr4 + nro + lane);
    v4f np1 = __builtin_nontemporal_load(pr4 + nro + 32 + lane);
    v4f ng0 = gt4[nro + lane];
    v4f ng1 = gt4[nro + 32 + lane];
    const int fro = ((h + 16) & 255) * (WW / 4);
    __builtin_prefetch(pr4 + fro + lane, 0, 1);   // global_prefetch_b8
    __builtin_prefetch(gt4 + fro + lane, 0, 1);

    float pm = fmaxf(fmaxf(fmaxf(p0.x, p0.y), fmaxf(p0.z, p0.w)),
                     fmaxf(fmaxf(p1.x, p1.y), fmaxf(p1.z, p1.w)));
    pm = wave_max32(pm);  // exact row max

    float gsq = g0.x * g0.x;
    gsq = fmaf(g0.y, g0.y, gsq);
    gsq = fmaf(g0.z, g0.z, gsq);
    gsq = fmaf(g0.w, g0.w, gsq);
    gsq = fmaf(g1.x, g1.x, gsq);
    gsq = fmaf(g1.y, g1.y, gsq);
    gsq = fmaf(g1.z, g1.z, gsq);
    gsq = fmaf(g1.w, g1.w, gsq);

    // branch-free so EXEC stays all-ones for the WMMA reductions
    float cross = 0.0f, cnt = 0.0f;
    cross += (p0.x == pm) ? g0.x : 0.0f;  cnt += (p0.x == pm) ? 1.0f : 0.0f;
    cross += (p0.y == pm) ? g0.y : 0.0f;  cnt += (p0.y == pm) ? 1.0f : 0.0f;
    cross += (p0.z == pm) ? g0.z : 0.0f;  cnt += (p0.z == pm) ? 1.0f : 0.0f;
    cross += (p0.w == pm) ? g0.w : 0.0f;  cnt += (p0.w == pm) ? 1.0f : 0.0f;
    cross += (p1.x == pm) ? g1.x : 0.0f;  cnt += (p1.x == pm) ? 1.0f : 0.0f;
    cross += (p1.y == pm) ? g1.y : 0.0f;  cnt += (p1.y == pm) ? 1.0f : 0.0f;
    cross += (p1.z == pm) ? g1.z : 0.0f;  cnt += (p1.z == pm) ? 1.0f : 0.0f;
    cross += (p1.w == pm) ? g1.w : 0.0f;  cnt += (p1.w == pm) ? 1.0f : 0.0f;

    float gs = wave_sum32(gsq);
    float cr = wave_sum32(cross);
    float cn = wave_sum32(cnt);
    if (lane == 0) {
      v4f st; st.x = pm; st.y = gs; st.z = cr; st.w = cn;
      rowstats[bj * HH + h] = st;
    }
    p0 = np0; p1 = np1; g0 = ng0; g1 = ng1;
  }
}

// -------- Kernel 1b: zero the mask accumulators (graph-safe init) ---------
__global__ void __launch_bounds__(256) k_init(float* __restrict__ maskSum) {
  const int i = blockIdx.x * 256 + threadIdx.x;
  if (i < BJ) maskSum[i] = 0.0f;
}

// -------- Kernel 2: gt_mask residue sums via the Tensor Data Mover --------
// reshape(B,-1,13).sum(1) = stride-13 residue sums over each b-slab: the one
// pattern that can't coalesce from VMEM.  TDM DMAs a contiguous 13312-float
// tile (1024 chunks x 13) into LDS (double-buffered, s_wait_tensorcnt),
// overlapping DMA with compute; y_gt is L2-resident after K1.  Each thread
// reads 13 consecutive floats per chunk: chunk base = 13*t words, gcd(13,64)=1
// -> 32 lanes hit 32 distinct banks, conflict-free.  acc[r] holds residue r
// exactly (chunk starts are multiples of 13).
#define TILE_F   13312              // floats per TDM tile (1024*13)
#define TILES_WG 8                  // tiles per workgroup
__global__ void __launch_bounds__(256) k_mask(const float* __restrict__ ygt,
                                              float* __restrict__ maskSum) {
  __shared__ float buf[2][TILE_F];
  __shared__ float part[8][13];
  const int b    = blockIdx.x >> 3;   // 8 WGs per batch element
  const int oct  = blockIdx.x & 7;
  const int tid  = threadIdx.x;
  const int lane = tid & 31;
  const int wv   = tid >> 5;
  const unsigned long long base =
      (unsigned long long)(uintptr_t)ygt +
      (unsigned long long)b * (JJ * HH * WW) * 4ull +
      (unsigned long long)oct * TILES_WG * TILE_F * 4ull;
  const unsigned int lds0 = (unsigned int)(uintptr_t)&buf[0][0];
  const unsigned int lds1 = (unsigned int)(uintptr_t)&buf[1][0];

  float acc[13];
#pragma unroll
  for (int r = 0; r < 13; ++r) acc[r] = 0.0f;

  if (wv == 0) tdm_load_1d(lds0, base, TILE_F);   // prologue: tile 0 -> buf0

  for (int t = 0; t < TILES_WG; ++t) {
    if (wv == 0) {
      if (t + 1 < TILES_WG) {
        tdm_load_1d((t & 1) ? lds0 : lds1,
                    base + (unsigned long long)(t + 1) * TILE_F * 4ull, TILE_F);
        __builtin_amdgcn_s_wait_tensorcnt(1);   // tile t landed
      } else {
        __builtin_amdgcn_s_wait_tensorcnt(0);
      }
    }
    __syncthreads();
    const float* cur = buf[t & 1];
#pragma unroll
    for (int k = 0; k < 4; ++k) {
      const float* p = cur + (tid + (k << 8)) * 13;
#pragma unroll
      for (int r = 0; r < 13; ++r) acc[r] += p[r];
    }
    __syncthreads();  // all waves done reading before buf is re-filled
  }

#pragma unroll
  for (int r = 0; r < 13; ++r) {
    float s = wave_sum32(acc[r]);               // WMMA wave reduction
    if (lane == 0) part[wv][r] = s;
  }
  __syncthreads();
  if (tid < 13) {
    float s = 0.0f;
#pragma unroll
    for (int w = 0; w < 8; ++w) s += part[w][tid];
    atomicAdd(&maskSum[b * JJ + tid], s);       // only the sign is consumed
  }
}

// -------- Kernel 3: per (b,j) global max over rowmaxes; assemble distances.
// dist^2 = gsq + (rowmax==mx ? cnt - 2*cross : 0); absent joints contribute
// nothing (NaN-excluded in the reference).
__global__ void __launch_bounds__(256) k_finalize(const v4f* __restrict__ rowstats,
                                                  const float* __restrict__ maskSum,
                                                  float* __restrict__ totals,
                                                  float* __restrict__ counts) {
  const int bj  = blockIdx.x;
  const int tid = threadIdx.x;
  v4f st = rowstats[bj * HH + tid];

  __shared__ float smx[8];
  float m = wave_max32(st.x);
  if ((tid & 31) == 0) smx[tid >> 5] = m;
  __syncthreads();
  float mx = smx[0];
#pragma unroll
  for (int i = 1; i < 8; ++i) mx = fmaxf(mx, smx[i]);

  const float present = (maskSum[bj] > 0.0f) ? 1.0f : 0.0f;
  float d2   = st.y + ((st.x == mx) ? (st.w - 2.0f * st.z) : 0.0f);
  float dist = sqrtf(fmaxf(d2, 0.0f));

  float ts = wave_sum32(present * dist);
  __shared__ float ssum[8];
  if ((tid & 31) == 0) ssum[tid >> 5] = ts;
  __syncthreads();
  if (tid == 0) {
    float T = 0.0f;
#pragma unroll
    for (int i = 0; i < 8; ++i) T += ssum[i];
    totals[bj] = T;
    counts[bj] = present * 256.0f;   // H rows per present joint
  }
}

// -------- Kernel 4: final nanmean scalar ----------------------------------
__global__ void __launch_bounds__(256) k_mean(const float* __restrict__ totals,
                                              const float* __restrict__ counts,
                                              float* __restrict__ out) {
  const int tid = threadIdx.x;
  float t = 0.0f, c = 0.0f;
  for (int i = tid; i < BJ; i += 256) { t += totals[i]; c += counts[i]; }
  t = wave_sum32(t);
  c = wave_sum32(c);
  __shared__ float st[8], sc[8];
  if ((tid & 31) == 0) { st[tid >> 5] = t; sc[tid >> 5] = c; }
  __syncthreads();
  if (tid == 0) {
    float T = 0.0f, C = 0.0f;
#pragma unroll
    for (int i = 0; i < 8; ++i) { T += st[i]; C += sc[i]; }
    out[0] = T / C;
  }
}

extern "C" void kernel_launch(void* const* d_in, const int* in_sizes, int n_in,
                              void* d_out, int out_size, void* d_ws, size_t ws_size,
                              hipStream_t stream) {
  const float* ypr = (const float*)d_in[0];
  const float* ygt = (const float*)d_in[1];
  float* ws = (float*)d_ws;
  // d_ws layout (floats): rowstats[416*256*4] | maskSum[416] | totals[416] | counts[416]
  v4f*   rowstats = (v4f*)ws;
  float* maskSum  = ws + (size_t)BJ * HH * 4;
  float* totals   = maskSum + BJ;
  float* counts   = totals + BJ;

  k_init    <<<2,   256, 0, stream>>>(maskSum);
  k_rowstats<<<BJ,  256, 0, stream>>>(ypr, ygt, rowstats);
  k_mask    <<<BB * 8, 256, 0, stream>>>(ygt, maskSum);
  k_finalize<<<BJ,  256, 0, stream>>>(rowstats, maskSum, totals, counts);
  k_mean    <<<1,   256, 0, stream>>>(totals, counts, (float*)d_out);
}